// GCN_2370821947943
// MI455X (gfx1250) — compile-verified
//
#include <hip/hip_runtime.h>
#include <hip/hip_bf16.h>
#include <math.h>

#define NFEAT 512
#define NHID  128
#define NCLS  40

typedef __attribute__((ext_vector_type(2))) float v2f;
typedef __attribute__((ext_vector_type(8))) float v8f;

// ---------------- degree / normalization ----------------
__global__ void k_deg_count(const int* __restrict__ dst, float* __restrict__ cnt,
                            long long E) {
  long long e = (long long)blockIdx.x * blockDim.x + threadIdx.x;
  if (e < E) atomicAdd(&cnt[dst[e]], 1.0f);
}

__global__ void k_dinv(float* __restrict__ cnt, int N) {
  int i = blockIdx.x * blockDim.x + threadIdx.x;
  if (i < N) cnt[i] = rsqrtf(cnt[i] + 1.0f);  // +1 = self loop; deg > 0 always
}

// ---------------- GEMM1: [N,512] x [512,128] -> [Npad,128], fp32 WMMA ----------------
// Block = 256 thr = 8 waves; block covers 64 rows x 128 cols.
// Each wave computes a 32x32 tile: 2 A-frags x 2 B-frags -> 4 WMMA per k-step.
// Output rows are padded in ws, so stores are unconditional (no exec juggling).
__global__ __launch_bounds__(256) void k_gemm1(const float* __restrict__ X,
                                               const float* __restrict__ W,
                                               float* __restrict__ H, int N) {
  const int wave = threadIdx.x >> 5;
  const int lane = threadIdx.x & 31;
  const int r    = lane & 15;   // A: row-in-tile, B/C: col-in-tile
  const int kh   = lane >> 4;   // half-wave -> K offset 0 / 2
  const int row0 = blockIdx.x * 64 + (wave >> 2) * 32;
  const int col0 = (wave & 3) * 32;

  int ar0 = row0 + r;      if (ar0 >= N) ar0 = N - 1;  // clamp: EXEC stays full
  int ar1 = row0 + 16 + r; if (ar1 >= N) ar1 = N - 1;
  const float* xr0 = X + (size_t)ar0 * NFEAT;
  const float* xr1 = X + (size_t)ar1 * NFEAT;
  const float* wc  = W + col0 + r;   // B cols [col0+r] and [col0+16+r]

  v8f acc00 = {}, acc01 = {}, acc10 = {}, acc11 = {};
#pragma unroll 2
  for (int k0 = 0; k0 < NFEAT; k0 += 4) {
    const int ka = k0 + (kh << 1);
    v2f a0, a1, b0, b1;
    a0.x = xr0[ka];  a0.y = xr0[ka + 1];
    a1.x = xr1[ka];  a1.y = xr1[ka + 1];
    b0.x = wc[(size_t)ka * NHID];        b0.y = wc[(size_t)(ka + 1) * NHID];
    b1.x = wc[(size_t)ka * NHID + 16];   b1.y = wc[(size_t)(ka + 1) * NHID + 16];
    acc00 = __builtin_amdgcn_wmma_f32_16x16x4_f32(false, a0, false, b0, (short)0, acc00, false, false);
    acc01 = __builtin_amdgcn_wmma_f32_16x16x4_f32(false, a0, false, b1, (short)0, acc01, false, false);
    acc10 = __builtin_amdgcn_wmma_f32_16x16x4_f32(false, a1, false, b0, (short)0, acc10, false, false);
    acc11 = __builtin_amdgcn_wmma_f32_16x16x4_f32(false, a1, false, b1, (short)0, acc11, false, false);
  }
  // C layout: VGPR j -> row (kh*8 + j), col = r (lanes 0-15) / r (lanes 16-31)
  float* o00 = H + (size_t)(row0 + (kh << 3)) * NHID + col0 + r;
  float* o10 = o00 + (size_t)16 * NHID;
#pragma unroll
  for (int j = 0; j < 8; ++j) {
    o00[(size_t)j * NHID]      = acc00[j];
    o00[(size_t)j * NHID + 16] = acc01[j];
    o10[(size_t)j * NHID]      = acc10[j];
    o10[(size_t)j * NHID + 16] = acc11[j];
  }
}

// ---------------- GEMM2: [N,128] x [128,40] -> [Npad,40], fp32 WMMA ----------------
// Block = 96 thr = 3 waves covering 48 cols (40 padded; pad cols read as 0).
// Each wave computes 32 rows x 16 cols: 2 A-frags share 1 B-frag -> 2 WMMA/k-step.
__global__ __launch_bounds__(96) void k_gemm2(const float* __restrict__ H1,
                                              const float* __restrict__ W,
                                              float* __restrict__ H2, int N) {
  const int wave = threadIdx.x >> 5;
  const int lane = threadIdx.x & 31;
  const int r    = lane & 15;
  const int kh   = lane >> 4;
  const int row0 = blockIdx.x * 32;
  const int col  = wave * 16 + r;
  const bool colok = col < NCLS;

  int ar0 = row0 + r;      if (ar0 >= N) ar0 = N - 1;
  int ar1 = row0 + 16 + r; if (ar1 >= N) ar1 = N - 1;
  const float* hr0 = H1 + (size_t)ar0 * NHID;
  const float* hr1 = H1 + (size_t)ar1 * NHID;

  v8f acc0 = {}, acc1 = {};
#pragma unroll 2
  for (int k0 = 0; k0 < NHID; k0 += 4) {
    const int ka = k0 + (kh << 1);
    v2f a0, a1, b;
    a0.x = hr0[ka];  a0.y = hr0[ka + 1];
    a1.x = hr1[ka];  a1.y = hr1[ka + 1];
    b.x = colok ? W[(size_t)ka * NCLS + col] : 0.0f;       // value-select, no divergence
    b.y = colok ? W[(size_t)(ka + 1) * NCLS + col] : 0.0f;
    acc0 = __builtin_amdgcn_wmma_f32_16x16x4_f32(false, a0, false, b, (short)0, acc0, false, false);
    acc1 = __builtin_amdgcn_wmma_f32_16x16x4_f32(false, a1, false, b, (short)0, acc1, false, false);
  }
  if (colok) {
    float* o0 = H2 + (size_t)(row0 + (kh << 3)) * NCLS + col;
    float* o1 = o0 + (size_t)16 * NCLS;
#pragma unroll
    for (int j = 0; j < 8; ++j) {
      o0[(size_t)j * NCLS] = acc0[j];
      o1[(size_t)j * NCLS] = acc1[j];
    }
  }
}

// ---------------- edge scatter, 128 feats: one wave per edge ----------------
__global__ __launch_bounds__(256) void k_scatter1(const int* __restrict__ src,
                                                  const int* __restrict__ dst,
                                                  const float* __restrict__ dinv,
                                                  const float* __restrict__ h0,
                                                  float* __restrict__ agg,
                                                  long long E) {
  const int lane = threadIdx.x & 31;
  const long long e = (long long)blockIdx.x * 8 + (threadIdx.x >> 5);
  if (e >= E) return;
  const int s = src[e], d = dst[e];
  const float w = dinv[s] * dinv[d];
  const float4 v = ((const float4*)(h0 + (size_t)s * NHID))[lane];
  float* out = agg + (size_t)d * NHID + lane * 4;
  atomicAdd(out + 0, v.x * w);
  atomicAdd(out + 1, v.y * w);
  atomicAdd(out + 2, v.z * w);
  atomicAdd(out + 3, v.w * w);
}

// ---------------- self-loop + bias + ReLU (agg1 -> h1 in place) ----------------
__global__ void k_relu(float* __restrict__ agg, const float* __restrict__ h0,
                       const float* __restrict__ dinv, const float* __restrict__ b1,
                       int N) {
  const long long idx = (long long)blockIdx.x * blockDim.x + threadIdx.x;
  if (idx >= (long long)N * NHID) return;
  const int i = (int)(idx >> 7);
  const int f = (int)(idx & (NHID - 1));
  const float di = dinv[i];
  float v = agg[idx] + h0[idx] * di * di + b1[f];
  agg[idx] = v > 0.0f ? v : 0.0f;
}

// ---------------- edge scatter, 40 feats ----------------
__global__ __launch_bounds__(256) void k_scatter2(const int* __restrict__ src,
                                                  const int* __restrict__ dst,
                                                  const float* __restrict__ dinv,
                                                  const float* __restrict__ h2,
                                                  float* __restrict__ agg,
                                                  long long E) {
  const int lane = threadIdx.x & 31;
  const long long e = (long long)blockIdx.x * 8 + (threadIdx.x >> 5);
  if (e >= E) return;
  const int s = src[e], d = dst[e];
  const float w = dinv[s] * dinv[d];
  const float* hs = h2 + (size_t)s * NCLS;
  float* out = agg + (size_t)d * NCLS;
  atomicAdd(out + lane, hs[lane] * w);                 // lanes 0..31 -> f 0..31
  if (lane < NCLS - 32) atomicAdd(out + 32 + lane, hs[32 + lane] * w);
}

// ---------------- self-loop + bias + log_softmax: one wave per node ----------------
__global__ __launch_bounds__(256) void k_final(const float* __restrict__ agg,
                                               const float* __restrict__ h2,
                                               const float* __restrict__ dinv,
                                               const float* __restrict__ b2,
                                               float* __restrict__ out, int N) {
  const int lane = threadIdx.x & 31;
  const int n = blockIdx.x * 8 + (threadIdx.x >> 5);
  if (n >= N) return;
  const float di = dinv[n];
  const float dd = di * di;
  const size_t base = (size_t)n * NCLS;
  const int f1 = lane + 32;
  const bool ok1 = f1 < NCLS;

  float l0 = agg[base + lane] + h2[base + lane] * dd + b2[lane];
  float l1 = ok1 ? (agg[base + f1] + h2[base + f1] * dd + b2[f1]) : -INFINITY;

  float m = fmaxf(l0, l1);
  for (int o = 16; o > 0; o >>= 1) m = fmaxf(m, __shfl_xor(m, o, 32));
  float s = expf(l0 - m) + (ok1 ? expf(l1 - m) : 0.0f);
  for (int o = 16; o > 0; o >>= 1) s += __shfl_xor(s, o, 32);
  const float lse = logf(s);

  const size_t NO = (size_t)N * NCLS;  // logits are 2nd output
  out[base + lane]      = l0 - m - lse;
  out[NO + base + lane] = l0;
  if (ok1) {
    out[base + f1]      = l1 - m - lse;
    out[NO + base + f1] = l1;
  }
}

extern "C" void kernel_launch(void* const* d_in, const int* in_sizes, int n_in,
                              void* d_out, int out_size, void* d_ws, size_t ws_size,
                              hipStream_t stream) {
  const float* x  = (const float*)d_in[0];
  const int*   ei = (const int*)d_in[1];   // edge_index [2,E], int32
  const float* W1 = (const float*)d_in[2];
  const float* b1 = (const float*)d_in[3];
  const float* W2 = (const float*)d_in[4];
  const float* b2 = (const float*)d_in[5];

  const int N = in_sizes[0] / NFEAT;
  const long long E = in_sizes[1] / 2;
  const int* src = ei;
  const int* dst = ei + E;

  const int Npad = (N + 63) & ~63;               // row padding for guard-free stores
  float* ws   = (float*)d_ws;
  float* dinv = ws;                              // Npad
  float* h0   = dinv + Npad;                     // Npad*128
  float* agg1 = h0 + (size_t)Npad * NHID;        // Npad*128 (becomes h1)
  float* h2   = agg1 + (size_t)Npad * NHID;      // Npad*40
  float* agg2 = h2 + (size_t)Npad * NCLS;        // Npad*40

  hipMemsetAsync(dinv, 0, sizeof(float) * (size_t)Npad, stream);
  hipMemsetAsync(agg1, 0, sizeof(float) * (size_t)Npad * NHID, stream);
  hipMemsetAsync(agg2, 0, sizeof(float) * (size_t)Npad * NCLS, stream);

  k_deg_count<<<(int)((E + 255) / 256), 256, 0, stream>>>(dst, dinv, E);
  k_dinv<<<(N + 255) / 256, 256, 0, stream>>>(dinv, N);

  k_gemm1<<<Npad / 64, 256, 0, stream>>>(x, W1, h0, N);
  k_scatter1<<<(int)((E + 7) / 8), 256, 0, stream>>>(src, dst, dinv, h0, agg1, E);
  k_relu<<<(int)(((long long)N * NHID + 255) / 256), 256, 0, stream>>>(agg1, h0, dinv, b1, N);

  k_gemm2<<<Npad / 32, 96, 0, stream>>>(agg1, W2, h2, N);
  k_scatter2<<<(int)((E + 7) / 8), 256, 0, stream>>>(src, dst, dinv, h2, agg2, E);
  k_final<<<(N + 7) / 8, 256, 0, stream>>>(agg2, h2, dinv, b2, (float*)d_out, N);
}